// Model_62955630625112
// MI455X (gfx1250) — compile-verified
//
#include <hip/hip_runtime.h>

#define FH 2048
#define FW 2048
#define BSZ 16
#define SR 16
#define NBX (FW / BSZ)   // 128
#define NBY (FH / BSZ)   // 128
// Tile handled per workgroup: 2 block-rows x 4 block-cols, 8 waves, 1 wave/block
#define TBY 2
#define TBX 4
#define REF_H (TBY * BSZ + 2 * SR)   // 64
#define REF_W (TBX * BSZ + 2 * SR)   // 96 (columns actually read)
#define REF_STRIDE 104               // pad: 2*104 % 64 == 16 -> lane halves hit disjoint banks

typedef float v2f __attribute__((ext_vector_type(2)));
typedef float v8f __attribute__((ext_vector_type(8)));
typedef unsigned int v4u __attribute__((ext_vector_type(4)));
typedef int v8i __attribute__((ext_vector_type(8)));
typedef int v4i __attribute__((ext_vector_type(4)));

#if __has_builtin(__builtin_amdgcn_tensor_load_to_lds) && __has_builtin(__builtin_amdgcn_s_wait_tensorcnt)
#define USE_TDM 1
#else
#define USE_TDM 0
#endif

// Sum all 16 values within each 16-lane row via DPP row-rotates (VALU pipe,
// avoids ds_bpermute traffic on the LDS pipe). After strides 1,2,4,8 every
// lane holds the row-wide sum.
__device__ __forceinline__ float rowsum16(float s) {
#if __has_builtin(__builtin_amdgcn_update_dpp)
  s += __int_as_float(__builtin_amdgcn_update_dpp(0, __float_as_int(s), 0x121, 0xF, 0xF, true)); // row_ror:1
  s += __int_as_float(__builtin_amdgcn_update_dpp(0, __float_as_int(s), 0x122, 0xF, 0xF, true)); // row_ror:2
  s += __int_as_float(__builtin_amdgcn_update_dpp(0, __float_as_int(s), 0x124, 0xF, 0xF, true)); // row_ror:4
  s += __int_as_float(__builtin_amdgcn_update_dpp(0, __float_as_int(s), 0x128, 0xF, 0xF, true)); // row_ror:8
#else
  s += __shfl_xor(s, 1);
  s += __shfl_xor(s, 2);
  s += __shfl_xor(s, 4);
  s += __shfl_xor(s, 8);
#endif
  return s;
}

__global__ __launch_bounds__(256) void me_sad_kernel(const float* __restrict__ cur,
                                                     const float* __restrict__ ref,
                                                     float* __restrict__ out) {
  __shared__ float refS[REF_H * REF_STRIDE];

  const int t = threadIdx.x;
  const int gx0 = blockIdx.x * (TBX * BSZ);   // tile origin in frame
  const int gy0 = blockIdx.y * (TBY * BSZ);
  const int startX = gx0 - SR;
  const int startY = gy0 - SR;
  // TDM path only for windows fully inside the frame (incl. the 8 unread
  // alignment columns) so D# OOB semantics never come into play.
  const bool interior = (startX >= 0) && (startY >= 0) &&
                        (startX + REF_STRIDE <= FW) && (startY + REF_H <= FH);
  (void)interior;

#if USE_TDM
  if (interior) {
    if (t < 32) {   // one wave drives the Tensor Data Mover
      unsigned long long ga =
          (unsigned long long)(const void*)(ref + (size_t)startY * FW + startX);
      unsigned int ldsb = (unsigned int)(unsigned long long)(void*)refS; // flat LDS addr low32 = LDS byte offset
      v4u g0;
      g0[0] = 1u;                                   // count=1 valid user descriptor
      g0[1] = ldsb;                                 // lds_addr (bytes)
      g0[2] = (unsigned)(ga & 0xFFFFFFFFu);         // global_addr[31:0]
      g0[3] = (unsigned)((ga >> 32) & 0x01FFFFFFu)  // global_addr[56:32]
              | 0x80000000u;                        // type = 2 ("image")
      v8i g1;
      g1[0] = 0x00020000;                           // data_size = 2 (4 bytes), no pad/iter/multicast
      g1[1] = (int)(2048u << 16);                   // tensor_dim0[15:0]  = 2048
      g1[2] = (int)(2048u << 16);                   // tensor_dim1[15:0]  = 2048
      g1[3] = (int)((unsigned)REF_STRIDE << 16);    // tile_dim0 = 104 floats (rows pack at stride 104)
      g1[4] = REF_H;                                // tile_dim1 = 64 rows, tile_dim2 = 0
      g1[5] = FW;                                   // tensor_dim0_stride = 2048
      g1[6] = 0;
      g1[7] = 0;
      v4i g2 = {0, 0, 0, 0};                        // tile_dim3=0 -> unused (2D)
      v4i g3 = {0, 0, 0, 0};
      v8i g4 = {0, 0, 0, 0, 0, 0, 0, 0};            // trailing group (unused for 2D)
      __builtin_amdgcn_tensor_load_to_lds(g0, g1, g2, g3, g4, 0);
      __builtin_amdgcn_s_wait_tensorcnt(0);
    }
  } else
#endif
  {
    // Manual staging with zero padding (frame edges, or no-TDM fallback)
#pragma unroll
    for (int i = 0; i < (REF_H * REF_W) / 256; ++i) {
      int idx = t + i * 256;
      int ry = idx / REF_W;
      int rx = idx - ry * REF_W;
      int gy = startY + ry;
      int gx = startX + rx;
      float v = 0.0f;
      if ((unsigned)gy < (unsigned)FH && (unsigned)gx < (unsigned)FW)
        v = ref[gy * FW + gx];
      refS[ry * REF_STRIDE + rx] = v;
    }
  }
  __syncthreads();

  const int wave = t >> 5;
  const int lane = t & 31;
  const int h = lane >> 4;   // half selector (WMMA B-layout lane group)
  const int n = lane & 15;   // column index N within the 16x16 block
  const int br = wave >> 2;  // block row in tile (0..1)
  const int bc = wave & 3;   // block col in tile (0..3)

  // This lane's 8 current-frame pixels in WMMA B layout:
  // chunk q (K = 4q..4q+3): lane holds rows k = 4q + 2h + v, column n
  float curv[4][2];
  {
    const int cy0 = gy0 + br * BSZ;
    const int cx0 = gx0 + bc * BSZ;
#pragma unroll
    for (int q = 0; q < 4; ++q)
#pragma unroll
      for (int v = 0; v < 2; ++v)
        curv[q][v] = cur[(cy0 + 4 * q + 2 * h + v) * FW + (cx0 + n)];
  }

  const v2f ones = {1.0f, 1.0f};   // A matrix = 16x4 of ones (column-sum operator)
  float best = INFINITY, bdx = 0.0f, bdy = 0.0f;

  const int rowBase = br * BSZ + SR;       // LDS row for block row 0 at dy=0
  const int colBase = bc * BSZ + SR + n;   // LDS col for this lane at dx=0

  for (int dy = -SR; dy <= SR; ++dy) {
    const float* rs = &refS[(rowBase + dy) * REF_STRIDE + colBase];
#pragma unroll 3
    for (int dx = -SR; dx <= SR; ++dx) {
      v8f acc = {0.0f, 0.0f, 0.0f, 0.0f, 0.0f, 0.0f, 0.0f, 0.0f};
      float part = 0.0f;
      (void)part;
#pragma unroll
      for (int q = 0; q < 4; ++q) {
        const int k0 = 4 * q + 2 * h;
        v2f b;
        b[0] = __builtin_fabsf(curv[q][0] - rs[(k0 + 0) * REF_STRIDE + dx]);
        b[1] = __builtin_fabsf(curv[q][1] - rs[(k0 + 1) * REF_STRIDE + dx]);
#if __has_builtin(__builtin_amdgcn_wmma_f32_16x16x4_f32)
        // D = ones(16x4) x |diff|(4x16) + C -> every D row = column sums.
        // 4 chained accumulations cover K = 0..15, co-executing with the
        // dual-issued VALU sub/abs stream.
        acc = __builtin_amdgcn_wmma_f32_16x16x4_f32(
            false, ones, false, b, (short)0, acc, false, false);
#else
        part += b[0] + b[1];
#endif
      }
#if __has_builtin(__builtin_amdgcn_wmma_f32_16x16x4_f32)
      // acc[0]: lane (l&15) holds colsum(N); lanes 16-31 hold the M=8 copy.
      // DPP rotate-reduce within each 16-lane row -> every lane has full SAD.
      float s = rowsum16(acc[0]);
#else
      float s = rowsum16(part);
      s += __shfl_xor(s, 16);
#endif
      // Strict '<' + in-order dy-outer/dx-inner scan == reference tie-break.
      bool better = s < best;
      best = better ? s : best;
      bdx = better ? (float)dx : bdx;
      bdy = better ? (float)dy : bdy;
    }
  }

  if (lane == 0) {
    const int by = blockIdx.y * TBY + br;
    const int bx = blockIdx.x * TBX + bc;
    const int o = by * NBX + bx;
    out[o] = bdx;                        // motion_x
    out[NBY * NBX + o] = bdy;            // motion_y
    out[2 * NBY * NBX + o] = best;       // min_sad
  }
}

extern "C" void kernel_launch(void* const* d_in, const int* in_sizes, int n_in,
                              void* d_out, int out_size, void* d_ws, size_t ws_size,
                              hipStream_t stream) {
  const float* cur = (const float*)d_in[0];   // current_frame
  const float* ref = (const float*)d_in[1];   // reference_frame
  float* out = (float*)d_out;                 // [motion_x | motion_y | min_sad]
  (void)in_sizes; (void)n_in; (void)out_size; (void)d_ws; (void)ws_size;

  dim3 grid(NBX / TBX, NBY / TBY);   // 32 x 64 tiles
  dim3 block(256);                   // 8 waves (wave32)
  me_sad_kernel<<<grid, block, 0, stream>>>(cur, ref, out);
}